// GCN_16217796510272
// MI455X (gfx1250) — compile-verified
//
#include <hip/hip_runtime.h>
#include <math.h>

typedef __attribute__((ext_vector_type(2))) float v2f;
typedef __attribute__((ext_vector_type(8))) float v8f;

#define Bn 32
#define Tn 243
#define Jn 17
#define Cn 256
#define Hn 64
#define NROWS (Bn*Tn*Jn)   // 132192
#define NFRAMES (Bn*Tn)    // 7776
#define BM 64
#define KC 16

// workspace layout (byte offsets)
#define WS_NBR   0      // int[34]
#define WS_WGT   192    // float[34]
#define WS_S     384    // float[17]
#define WS_SUM   512    // float[17]
#define WS_SQ    640    // float[17]
#define WS_SCALE 768    // float[17]
#define WS_SHIFT 896    // float[17]
#define WS_H     1024   // float[NROWS*Cn]  (~135 MB)

// ---------------------------------------------------------------------------
// Kernel 1: normalized adjacency -> per-joint neighbor lists / weights / rowsum
// ---------------------------------------------------------------------------
__global__ void gcn_prep_kernel(const float* __restrict__ adj, void* ws) {
  if (threadIdx.x != 0) return;
  int*   nbr  = (int*)  ((char*)ws + WS_NBR);
  float* wgt  = (float*)((char*)ws + WS_WGT);
  float* svec = (float*)((char*)ws + WS_S);
  float* sums = (float*)((char*)ws + WS_SUM);
  float* sq   = (float*)((char*)ws + WS_SQ);
  float dinv[Jn];
  for (int i = 0; i < Jn; ++i) {
    float d = 0.f;
    for (int j = 0; j < Jn; ++j) d += adj[i*Jn + j];
    dinv[i] = rsqrtf(d);
  }
  for (int i = 0; i < Jn; ++i) {
    int cnt = 0; float s = 0.f;
    for (int j = 0; j < Jn; ++j) {
      float a = adj[i*Jn + j];
      if (a != 0.f) {
        float w = dinv[i] * a * dinv[j];
        s += w;
        if (cnt < 2) { nbr[i*2 + cnt] = j; wgt[i*2 + cnt] = w; ++cnt; }
      }
    }
    for (; cnt < 2; ++cnt) { nbr[i*2 + cnt] = 0; wgt[i*2 + cnt] = 0.f; }
    svec[i] = s;
    sums[i] = 0.f; sq[i] = 0.f;
  }
}

// ---------------------------------------------------------------------------
// Kernel 2: fused dual GEMM with WMMA f32 16x16x4, software-pipelined.
//   h[row,:] = (sum_j A[i,j] x[bt,j,:]) @ V^T + x[row,:] @ U^T + U_b + s_j*V_b
// Block: 64 rows x 256 cols, 8 waves (2M x 4N), wave tile 32x64.
// ---------------------------------------------------------------------------
__global__ __launch_bounds__(256) void gcn_gemm_kernel(
    const float* __restrict__ x,  const float* __restrict__ Uw,
    const float* __restrict__ Ub, const float* __restrict__ Vw,
    const float* __restrict__ Vb, void* ws) {
  __shared__ float ldsAx[BM][KC + 1];   // pitch 17: conflict-free A frag reads
  __shared__ float ldsAg[BM][KC + 1];
  __shared__ float ldsBu[KC][272];      // pitch 272 = 16*17
  __shared__ float ldsBv[KC][272];
  __shared__ int   s_nbr[Jn * 2];
  __shared__ float s_wgt[Jn * 2];
  __shared__ float s_s[Jn];

  const int tid = threadIdx.x;
  const int r0  = blockIdx.x * BM;
  float* hbuf = (float*)((char*)ws + WS_H);

  if (tid < Jn * 2) {
    s_nbr[tid] = ((const int*)  ((const char*)ws + WS_NBR))[tid];
    s_wgt[tid] = ((const float*)((const char*)ws + WS_WGT))[tid];
  }
  if (tid < Jn) s_s[tid] = ((const float*)((const char*)ws + WS_S))[tid];
  __syncthreads();

  // Per-thread A-tile row bookkeeping: thread loads k-col (tid&15) of 4 rows.
  const int kk  = tid & 15;
  const int rr0 = tid >> 4;                 // 0..15
  long xbS[4], xb0[4], xb1[4];
  float w0r[4], w1r[4];
  bool valid[4];
  for (int p = 0; p < 4; ++p) {
    int row = r0 + rr0 + p * 16;
    valid[p] = (row < NROWS);
    int rowc = valid[p] ? row : 0;
    int bt = rowc / Jn;
    int j  = rowc - bt * Jn;
    long fb = (long)bt * Jn * Cn;
    xbS[p] = (long)rowc * Cn;
    xb0[p] = fb + (long)s_nbr[j*2 + 0] * Cn;
    xb1[p] = fb + (long)s_nbr[j*2 + 1] * Cn;
    w0r[p] = s_wgt[j*2 + 0];
    w1r[p] = s_wgt[j*2 + 1];
  }

  const int lane = tid & 31;
  const int wv = tid >> 5;     // 0..7
  const int wm = wv >> 2;      // 0..1
  const int wn = wv & 3;       // 0..3
  const int mt = wm * 32;
  const int nt = wn * 64;
  const int lm = lane & 15;
  const int lh = lane >> 4;    // 0: K=k..k+1 / M=v ; 1: K=k+2..k+3 / M=v+8

  v8f acc[2][4] = {};

  // ---- software pipeline: prefetch chunk into registers, then LDS ----
  float pAx[4], pA0[4], pA1[4];
  float4 pBu[4], pBv[4];

  // prologue: prefetch chunk 0
  {
    const int kc = 0;
#pragma unroll
    for (int p = 0; p < 4; ++p) {
      pAx[p] = valid[p] ? x[xbS[p] + kc + kk] : 0.f;
      pA0[p] = valid[p] ? x[xb0[p] + kc + kk] : 0.f;
      pA1[p] = valid[p] ? x[xb1[p] + kc + kk] : 0.f;
    }
#pragma unroll
    for (int q = 0; q < 4; ++q) {
      pBu[q] = *(const float4*)(Uw + (long)tid * Cn + kc + q * 4);
      pBv[q] = *(const float4*)(Vw + (long)tid * Cn + kc + q * 4);
    }
  }

  for (int kc = 0; kc < Cn; kc += KC) {
    // Commit prefetched chunk to LDS.
#pragma unroll
    for (int p = 0; p < 4; ++p) {
      int rl = rr0 + p * 16;
      ldsAx[rl][kk] = pAx[p];
      ldsAg[rl][kk] = fmaf(w0r[p], pA0[p], w1r[p] * pA1[p]);
    }
#pragma unroll
    for (int q = 0; q < 4; ++q) {
      ldsBu[q*4 + 0][tid] = pBu[q].x; ldsBu[q*4 + 1][tid] = pBu[q].y;
      ldsBu[q*4 + 2][tid] = pBu[q].z; ldsBu[q*4 + 3][tid] = pBu[q].w;
      ldsBv[q*4 + 0][tid] = pBv[q].x; ldsBv[q*4 + 1][tid] = pBv[q].y;
      ldsBv[q*4 + 2][tid] = pBv[q].z; ldsBv[q*4 + 3][tid] = pBv[q].w;
    }
    __syncthreads();

    // Issue next chunk's global loads; latency overlaps WMMA compute below.
    if (kc + KC < Cn) {
      const int kn = kc + KC;
#pragma unroll
      for (int p = 0; p < 4; ++p) {
        pAx[p] = valid[p] ? x[xbS[p] + kn + kk] : 0.f;
        pA0[p] = valid[p] ? x[xb0[p] + kn + kk] : 0.f;
        pA1[p] = valid[p] ? x[xb1[p] + kn + kk] : 0.f;
      }
#pragma unroll
      for (int q = 0; q < 4; ++q) {
        pBu[q] = *(const float4*)(Uw + (long)tid * Cn + kn + q * 4);
        pBv[q] = *(const float4*)(Vw + (long)tid * Cn + kn + q * 4);
      }
    }

#pragma unroll
    for (int k4 = 0; k4 < KC; k4 += 4) {
      const int kb = k4 + lh * 2;
      v2f ax[2], ag[2];
#pragma unroll
      for (int mi = 0; mi < 2; ++mi) {
        int rl = mt + mi * 16 + lm;
        ax[mi].x = ldsAx[rl][kb];     ax[mi].y = ldsAx[rl][kb + 1];
        ag[mi].x = ldsAg[rl][kb];     ag[mi].y = ldsAg[rl][kb + 1];
      }
      v2f bu[4], bv[4];
#pragma unroll
      for (int ni = 0; ni < 4; ++ni) {
        int n = nt + ni * 16 + lm;
        bu[ni].x = ldsBu[kb][n];      bu[ni].y = ldsBu[kb + 1][n];
        bv[ni].x = ldsBv[kb][n];      bv[ni].y = ldsBv[kb + 1][n];
      }
#pragma unroll
      for (int mi = 0; mi < 2; ++mi)
#pragma unroll
        for (int ni = 0; ni < 4; ++ni) {
          acc[mi][ni] = __builtin_amdgcn_wmma_f32_16x16x4_f32(
              false, ax[mi], false, bu[ni], (short)0, acc[mi][ni], false, false);
          acc[mi][ni] = __builtin_amdgcn_wmma_f32_16x16x4_f32(
              false, ag[mi], false, bv[ni], (short)0, acc[mi][ni], false, false);
        }
    }
    __syncthreads();
  }

  // Epilogue: + U_b + s_j * V_b, write h to workspace.
#pragma unroll
  for (int mi = 0; mi < 2; ++mi) {
    const int rbase = r0 + mt + mi * 16 + lh * 8;   // 8 consecutive rows
    const int jbase = rbase % Jn;                   // one modulo per mi
#pragma unroll
    for (int ni = 0; ni < 4; ++ni) {
      int n = nt + ni * 16 + lm;
      float ub = Ub[n], vb = Vb[n];
#pragma unroll
      for (int v = 0; v < 8; ++v) {
        int row = rbase + v;
        if (row < NROWS) {
          int j = jbase + v; if (j >= Jn) j -= Jn;
          hbuf[(long)row * Cn + n] = acc[mi][ni][v] + ub + s_s[j] * vb;
        }
      }
    }
  }
}

// ---------------------------------------------------------------------------
// Kernel 3: per-joint BN statistics (sum, sum of squares over B*T*C)
// ---------------------------------------------------------------------------
__global__ __launch_bounds__(256) void gcn_stats_kernel(void* ws) {
  __shared__ float ls[Jn], lq[Jn];
  const int tid = threadIdx.x;
  if (tid < Jn) { ls[tid] = 0.f; lq[tid] = 0.f; }
  __syncthreads();
  int row = blockIdx.x * 256 + tid;
  if (row < NROWS) {
    const float* p = (const float*)((const char*)ws + WS_H) + (long)row * Cn;
    float s = 0.f, q = 0.f;
    for (int c = 0; c < Cn; c += 4) {
      float4 v = *(const float4*)(p + c);
      s += v.x + v.y + v.z + v.w;
      q += v.x*v.x + v.y*v.y + v.z*v.z + v.w*v.w;
    }
    int j = row % Jn;
    atomicAdd(&ls[j], s);
    atomicAdd(&lq[j], q);
  }
  __syncthreads();
  if (tid < Jn) {
    atomicAdd((float*)((char*)ws + WS_SUM) + tid, ls[tid]);
    atomicAdd((float*)((char*)ws + WS_SQ)  + tid, lq[tid]);
  }
}

// ---------------------------------------------------------------------------
// Kernel 4: finalize BN scale/shift per joint
// ---------------------------------------------------------------------------
__global__ void gcn_finalize_kernel(const float* __restrict__ gamma,
                                    const float* __restrict__ beta, void* ws) {
  int j = threadIdx.x;
  if (j >= Jn) return;
  const float cnt = (float)((long)NFRAMES * Cn);
  float s = ((float*)((char*)ws + WS_SUM))[j];
  float q = ((float*)((char*)ws + WS_SQ))[j];
  float mean = s / cnt;
  float var  = q / cnt - mean * mean;     // biased, torch-BN style
  float sc = gamma[j] * rsqrtf(var + 1e-5f);
  ((float*)((char*)ws + WS_SCALE))[j] = sc;
  ((float*)((char*)ws + WS_SHIFT))[j] = beta[j] - mean * sc;
}

// ---------------------------------------------------------------------------
// Kernel 5: y = relu(x + BN(h)); attention gate (256->64->1); out = y * att
// One wave per row; 8 rows per 256-thread block. NROWS % 8 == 0.
// ---------------------------------------------------------------------------
__global__ __launch_bounds__(256) void gcn_bnatt_kernel(
    const float* __restrict__ x, const void* ws,
    const float* __restrict__ a1w, const float* __restrict__ a1b,
    const float* __restrict__ a2w, const float* __restrict__ a2b,
    float* __restrict__ out) {
  __shared__ float ly[8][Cn];
  const int tid  = threadIdx.x;
  const int lane = tid & 31;
  const int wv   = tid >> 5;
  const int row  = blockIdx.x * 8 + wv;
  const float* hbuf = (const float*)((const char*)ws + WS_H);
  const int j = row % Jn;
  const float sc = ((const float*)((const char*)ws + WS_SCALE))[j];
  const float sh = ((const float*)((const char*)ws + WS_SHIFT))[j];
  const long base = (long)row * Cn;

  float yv[8];
#pragma unroll
  for (int i = 0; i < 8; ++i) {
    int c = i * 32 + lane;
    float v = x[base + c] + sc * hbuf[base + c] + sh;
    v = fmaxf(v, 0.f);
    yv[i] = v;
    ly[wv][c] = v;
  }
  __syncthreads();

  // Two hidden units per lane: u = lane, lane+32.  float4 weight loads.
  float h0 = a1b[lane], h1 = a1b[lane + 32];
  const float* w0p = a1w + (long)lane * Cn;
  const float* w1p = a1w + (long)(lane + 32) * Cn;
#pragma unroll 4
  for (int c = 0; c < Cn; c += 4) {
    float4 w0 = *(const float4*)(w0p + c);
    float4 w1 = *(const float4*)(w1p + c);
    float y0 = ly[wv][c + 0], y1 = ly[wv][c + 1];
    float y2 = ly[wv][c + 2], y3 = ly[wv][c + 3];
    h0 = fmaf(w0.x, y0, h0); h0 = fmaf(w0.y, y1, h0);
    h0 = fmaf(w0.z, y2, h0); h0 = fmaf(w0.w, y3, h0);
    h1 = fmaf(w1.x, y0, h1); h1 = fmaf(w1.y, y1, h1);
    h1 = fmaf(w1.z, y2, h1); h1 = fmaf(w1.w, y3, h1);
  }
  h0 = fmaxf(h0, 0.f);
  h1 = fmaxf(h1, 0.f);
  float part = a2w[lane] * h0 + a2w[lane + 32] * h1;
  for (int off = 16; off > 0; off >>= 1)
    part += __shfl_xor(part, off, 32);
  float att = 1.f / (1.f + __expf(-(part + a2b[0])));

#pragma unroll
  for (int i = 0; i < 8; ++i) {
    int c = i * 32 + lane;
    out[base + c] = yv[i] * att;
  }
}

// ---------------------------------------------------------------------------
extern "C" void kernel_launch(void* const* d_in, const int* in_sizes, int n_in,
                              void* d_out, int out_size, void* d_ws, size_t ws_size,
                              hipStream_t stream) {
  const float* x   = (const float*)d_in[0];
  const float* adj = (const float*)d_in[1];
  const float* Uw  = (const float*)d_in[2];
  const float* Ub  = (const float*)d_in[3];
  const float* Vw  = (const float*)d_in[4];
  const float* Vb  = (const float*)d_in[5];
  const float* gam = (const float*)d_in[6];
  const float* bet = (const float*)d_in[7];
  const float* a1w = (const float*)d_in[8];
  const float* a1b = (const float*)d_in[9];
  const float* a2w = (const float*)d_in[10];
  const float* a2b = (const float*)d_in[11];
  float* out = (float*)d_out;
  (void)in_sizes; (void)n_in; (void)out_size; (void)ws_size;

  gcn_prep_kernel<<<1, 32, 0, stream>>>(adj, d_ws);
  gcn_gemm_kernel<<<(NROWS + BM - 1) / BM, 256, 0, stream>>>(x, Uw, Ub, Vw, Vb, d_ws);
  gcn_stats_kernel<<<(NROWS + 255) / 256, 256, 0, stream>>>(d_ws);
  gcn_finalize_kernel<<<1, 32, 0, stream>>>(gam, bet, d_ws);
  gcn_bnatt_kernel<<<NROWS / 8, 256, 0, stream>>>(x, d_ws, a1w, a1b, a2w, a2b, out);
}